// ManifoldMatchingLoss_61667140436172
// MI455X (gfx1250) — compile-verified
//
#include <hip/hip_runtime.h>
#include <hip/hip_bf16.h>

// CDNA5 / gfx1250 wave32 WMMA types
typedef float v2f __attribute__((ext_vector_type(2)));
typedef float v8f __attribute__((ext_vector_type(8)));
// pointee type expected by the async global->LDS builtin (from clang diagnostic)
typedef int v4i __attribute__((vector_size(4 * sizeof(int))));

#define BSZ    256    // batch size (rows of each Gram matrix)
#define SPLITK 16     // split-K factor (deterministic two-pass reduction)
#define KB     32     // K-block staged through LDS per iteration
#define BM     64     // output tile rows per workgroup
#define BN     64     // output tile cols per workgroup
#define LDA    36     // padded LDS row stride (floats): 16B-aligned rows, conflict-free
#define STAGE_OPS 8   // async b128 ops per wave per stage (4 A + 4 B chunks / lane)

#if defined(__has_builtin)
#  if __has_builtin(__builtin_amdgcn_global_load_async_to_lds_b128)
#    define HAVE_ASYNC_BUILTIN 1
#  else
#    define HAVE_ASYNC_BUILTIN 0
#  endif
#  if __has_builtin(__builtin_amdgcn_s_wait_asynccnt)
#    define HAVE_WAIT_BUILTIN 1
#  else
#    define HAVE_WAIT_BUILTIN 0
#  endif
#else
#  define HAVE_ASYNC_BUILTIN 0
#  define HAVE_WAIT_BUILTIN 0
#endif

// Async global->LDS 16B copy (GLOBAL_LOAD_ASYNC_TO_LDS_B128, ASYNCcnt-tracked).
// LDS operand: generic shared VA low 32 bits == group-segment offset (ISA 10.2
// aperture rules), so an int round-trip builds a valid AS(3) operand.
__device__ __forceinline__ void async_cp16(const float* g, const float* l) {
#if HAVE_ASYNC_BUILTIN
    __builtin_amdgcn_global_load_async_to_lds_b128(
        (__attribute__((address_space(1))) v4i*)(uintptr_t)g,
        (__attribute__((address_space(3))) v4i*)(uint32_t)(uintptr_t)l,
        0, 0);
#else
    asm volatile("global_load_async_to_lds_b128 %0, %1, off"
                 :: "v"((uint32_t)(uintptr_t)l),
                    "v"((unsigned long long)(uintptr_t)g)
                 : "memory");
#endif
}

__device__ __forceinline__ void wait_async_stage() {   // <= one stage in flight
#if HAVE_WAIT_BUILTIN
    __builtin_amdgcn_s_wait_asynccnt(STAGE_OPS);
#else
    asm volatile("s_wait_asynccnt 0x8" ::: "memory");
#endif
}
__device__ __forceinline__ void wait_async_zero() {
#if HAVE_WAIT_BUILTIN
    __builtin_amdgcn_s_wait_asynccnt(0);
#else
    asm volatile("s_wait_asynccnt 0x0" ::: "memory");
#endif
}

// Issue one K-block stage (64x32 A rows + 64x32 B rows) as async b128 copies.
__device__ __forceinline__ void issue_stage(const float* __restrict__ X, int K,
                                            int mBase, int nBase, long k0,
                                            float* Abuf, float* Bbuf, int lr, int lc) {
#pragma unroll
    for (int p = 0; p < 4; ++p) {
        const int r = p * 16 + lr;
        async_cp16(X + (size_t)(mBase + r) * K + k0 + lc, &Abuf[r * LDA + lc]);
        async_cp16(X + (size_t)(nBase + r) * K + k0 + lc, &Bbuf[r * LDA + lc]);
    }
}

// ---------------------------------------------------------------------------
// Fused Gram partial kernel for BOTH modalities (grid.z selects matrix+split).
// 64x64 tile per 4-wave workgroup; V_WMMA_F32_16X16X4_F32 (f32-exact);
// double-buffered LDS fed by async global->LDS copies.
// ---------------------------------------------------------------------------
__global__ __launch_bounds__(128)
void gram_partial_kernel(const float* __restrict__ Xf, int Kf, int kbfPer,
                         float* __restrict__ Gpf_,
                         const float* __restrict__ Xg, int Kg, int kbgPer,
                         float* __restrict__ Gpg_) {
    __shared__ float As[2][BM * LDA];
    __shared__ float Bs[2][BN * LDA];

    const float* X; int K, kbPerSplit, split; float* Gpart;
    if (blockIdx.z < SPLITK) { X = Xf; K = Kf; kbPerSplit = kbfPer; Gpart = Gpf_; split = blockIdx.z; }
    else                     { X = Xg; K = Kg; kbPerSplit = kbgPer; Gpart = Gpg_; split = blockIdx.z - SPLITK; }

    const int t    = threadIdx.x;     // 0..127
    const int wave = t >> 5;          // 0..3
    const int lane = t & 31;
    const int half = lane >> 4;       // 0: lanes 0-15, 1: lanes 16-31
    const int mr   = lane & 15;

    const int  nBase  = blockIdx.x * BN;
    const int  mBase  = blockIdx.y * BM;
    const long kStart = (long)split * kbPerSplit * KB;

    // cooperative-copy mapping: 128 threads cover 16 rows x 32 cols per pass
    const int lr = t >> 3;            // row within 16-row pass
    const int lc = (t & 7) << 2;      // col 0..28 step 4 (16B chunks)

    v8f acc0 = {}, acc1 = {}, acc2 = {}, acc3 = {};

    issue_stage(X, K, mBase, nBase, kStart, As[0], Bs[0], lr, lc);

    for (int kb = 0; kb < kbPerSplit; ++kb) {
        const int cur = kb & 1;
        if (kb + 1 < kbPerSplit) {
            issue_stage(X, K, mBase, nBase, kStart + (long)(kb + 1) * KB,
                        As[cur ^ 1], Bs[cur ^ 1], lr, lc);
            wait_async_stage();       // stage kb complete; stage kb+1 in flight
        } else {
            wait_async_zero();
        }
        __syncthreads();              // all waves' stage-kb LDS writes visible

        const float* Ac = As[cur];
        const float* Bc = Bs[cur];
        // 16x16x4 f32 fragments:
        //   A: lane(0-15) holds M=lane; VGPR0=K(2*half), VGPR1=K(2*half+1)
        //   B mirrors A (B = X^T: column n == row n of X)
#pragma unroll
        for (int kk = 0; kk < KB; kk += 4) {
            const int kc = kk + (half << 1);
            const v2f a  = *(const v2f*)&Ac[(wave * 16 + mr) * LDA + kc];
            const v2f b0 = *(const v2f*)&Bc[( 0 + mr) * LDA + kc];
            const v2f b1 = *(const v2f*)&Bc[(16 + mr) * LDA + kc];
            const v2f b2 = *(const v2f*)&Bc[(32 + mr) * LDA + kc];
            const v2f b3 = *(const v2f*)&Bc[(48 + mr) * LDA + kc];
            acc0 = __builtin_amdgcn_wmma_f32_16x16x4_f32(false, a, false, b0, (short)0, acc0, false, false);
            acc1 = __builtin_amdgcn_wmma_f32_16x16x4_f32(false, a, false, b1, (short)0, acc1, false, false);
            acc2 = __builtin_amdgcn_wmma_f32_16x16x4_f32(false, a, false, b2, (short)0, acc2, false, false);
            acc3 = __builtin_amdgcn_wmma_f32_16x16x4_f32(false, a, false, b3, (short)0, acc3, false, false);
        }
        __syncthreads();              // readers done before buffer is re-filled
    }

    // C/D layout: VGPR r -> M = r + 8*half, N = lane&15
    float* Gs = Gpart + (size_t)split * BSZ * BSZ;
#pragma unroll
    for (int r = 0; r < 8; ++r) {
        const size_t row = (size_t)(mBase + wave * 16 + r + 8 * half);
        Gs[row * BSZ + nBase +  0 + mr] = acc0[r];
        Gs[row * BSZ + nBase + 16 + mr] = acc1[r];
        Gs[row * BSZ + nBase + 32 + mr] = acc2[r];
        Gs[row * BSZ + nBase + 48 + mr] = acc3[r];
    }
}

// ---------------------------------------------------------------------------
// Exact f32 row squared norms for both modalities (blockIdx.y selects matrix),
// deterministic LDS tree reduction.
// ---------------------------------------------------------------------------
__global__ __launch_bounds__(256)
void rownorm_kernel(const float* __restrict__ Xf, int Kf, float* __restrict__ sqf,
                    const float* __restrict__ Xg, int Kg, float* __restrict__ sqg) {
    __shared__ float red[256];
    const float* X; int K; float* sq;
    if (blockIdx.y == 0) { X = Xf; K = Kf; sq = sqf; }
    else                 { X = Xg; K = Kg; sq = sqg; }
    const int row = blockIdx.x, t = threadIdx.x;
    const float* xr = X + (size_t)row * K;
    float s = 0.f;
    for (int k = t; k < K; k += 256) { const float v = xr[k]; s += v * v; }
    red[t] = s; __syncthreads();
    for (int st = 128; st > 0; st >>= 1) {
        if (t < st) red[t] += red[t + st];
        __syncthreads();
    }
    if (t == 0) sq[row] = red[0];
}

// ---------------------------------------------------------------------------
// Per-row: reduce split-K partials, z = -sqrt(max(d^2,EPS))/T, max-subtracted
// log-softmax for both modalities, KL row contribution sum_j A_g(logA_g-logA_f).
// ---------------------------------------------------------------------------
__global__ __launch_bounds__(256)
void kl_row_kernel(const float* __restrict__ Gpf, const float* __restrict__ Gpg,
                   const float* __restrict__ sqf, const float* __restrict__ sqg,
                   const float* __restrict__ temp, float* __restrict__ klrow) {
    __shared__ float red[256];
    const int i = blockIdx.x, j = threadIdx.x;

    float gf = 0.f, gg = 0.f;
    for (int s = 0; s < SPLITK; ++s) {
        gf += Gpf[((size_t)s * BSZ + i) * BSZ + j];
        gg += Gpg[((size_t)s * BSZ + i) * BSZ + j];
    }
    const float T = temp[0];
    const float d2f = sqf[i] + sqf[j] - 2.f * gf;
    const float d2g = sqg[i] + sqg[j] - 2.f * gg;
    const float zf = -sqrtf(fmaxf(d2f, 1e-12f)) / T;
    const float zg = -sqrtf(fmaxf(d2g, 1e-12f)) / T;

    red[j] = zf; __syncthreads();
    for (int st = 128; st > 0; st >>= 1) { if (j < st) red[j] = fmaxf(red[j], red[j + st]); __syncthreads(); }
    const float mf = red[0]; __syncthreads();
    red[j] = expf(zf - mf); __syncthreads();
    for (int st = 128; st > 0; st >>= 1) { if (j < st) red[j] += red[j + st]; __syncthreads(); }
    const float lsef = mf + logf(red[0]); __syncthreads();

    red[j] = zg; __syncthreads();
    for (int st = 128; st > 0; st >>= 1) { if (j < st) red[j] = fmaxf(red[j], red[j + st]); __syncthreads(); }
    const float mg = red[0]; __syncthreads();
    red[j] = expf(zg - mg); __syncthreads();
    for (int st = 128; st > 0; st >>= 1) { if (j < st) red[j] += red[j + st]; __syncthreads(); }
    const float lseg = mg + logf(red[0]); __syncthreads();

    const float lAf = zf - lsef;
    const float lAg = zg - lseg;
    red[j] = expf(lAg) * (lAg - lAf); __syncthreads();
    for (int st = 128; st > 0; st >>= 1) { if (j < st) red[j] += red[j + st]; __syncthreads(); }
    if (j == 0) klrow[i] = red[0];
}

__global__ __launch_bounds__(256)
void final_kernel(const float* __restrict__ klrow, float* __restrict__ out) {
    __shared__ float red[256];
    const int j = threadIdx.x;
    red[j] = klrow[j]; __syncthreads();
    for (int st = 128; st > 0; st >>= 1) { if (j < st) red[j] += red[j + st]; __syncthreads(); }
    if (j == 0) out[0] = red[0] / (float)BSZ;
}

// ---------------------------------------------------------------------------
extern "C" void kernel_launch(void* const* d_in, const int* in_sizes, int n_in,
                              void* d_out, int out_size, void* d_ws, size_t ws_size,
                              hipStream_t stream) {
    const float* img  = (const float*)d_in[0];   // [256, 196, 512] f32
    const float* lng  = (const float*)d_in[1];   // [256, 77, 512]  f32
    const float* temp = (const float*)d_in[2];   // [1] f32
    float* out = (float*)d_out;

    const int Kf = in_sizes[0] / BSZ;            // 100352 = 16*196*32
    const int Kg = in_sizes[1] / BSZ;            // 39424  = 16*77*32

    // workspace layout (f32): 2 x split-K partial Grams + norms + row KL
    float* Gpf   = (float*)d_ws;
    float* Gpg   = Gpf + (size_t)SPLITK * BSZ * BSZ;
    float* sqf   = Gpg + (size_t)SPLITK * BSZ * BSZ;
    float* sqg   = sqf + BSZ;
    float* klrow = sqg + BSZ;

    // both Gram matrices in one launch: grid.z in [0,16) image, [16,32) lang
    const dim3 grid(BSZ / BN, BSZ / BM, 2 * SPLITK);
    gram_partial_kernel<<<grid, dim3(128), 0, stream>>>(
        img, Kf, (Kf / KB) / SPLITK, Gpf,
        lng, Kg, (Kg / KB) / SPLITK, Gpg);
    rownorm_kernel<<<dim3(BSZ, 2), dim3(256), 0, stream>>>(img, Kf, sqf, lng, Kg, sqg);
    kl_row_kernel<<<BSZ, 256, 0, stream>>>(Gpf, Gpg, sqf, sqg, temp, klrow);
    final_kernel<<<1, 256, 0, stream>>>(klrow, out);
}